// ActionPrompt_84464826843799
// MI455X (gfx1250) — compile-verified
//
#include <hip/hip_runtime.h>

// out = (act @ Wv2 + bv2) @ Wo2 + bo2   (exact reduction of the reference:
// both softmaxes are over singleton axes -> identically 1.0, and the mean
// over combs of a comb-independent tensor is the tensor itself).

typedef float v2f __attribute__((ext_vector_type(2)));
typedef float v8f __attribute__((ext_vector_type(8)));

#define DIM    1024
#define A_ROWS 50
#define M_PAD  64

// Build zero-padded activation matrix Xp[64][1024] (rows >= 50 are zero).
__global__ void actionprompt_pad_kernel(const float* __restrict__ act,
                                        float* __restrict__ xp) {
    int idx = blockIdx.x * blockDim.x + threadIdx.x;   // 0..65535
    int row = idx >> 10;
    xp[idx] = (row < A_ROWS) ? act[idx] : 0.0f;        // act is dense [50,1024]
}

// Y[m_store x 1024] = X[64 x 1024] @ W[1024 x 1024] + bias[1024]
// One wave per 16-wide N-tile; each wave accumulates the full 64-row strip
// (4 M-tiles) so every 4x16 weight fragment is loaded exactly once.
__global__ __launch_bounds__(256) void actionprompt_gemm_wmma_f32(
    const float* __restrict__ X, const float* __restrict__ W,
    const float* __restrict__ bias, float* __restrict__ Y, int m_store)
{
    const int wave = (blockIdx.x * blockDim.x + threadIdx.x) >> 5; // 0..63
    const int lane = threadIdx.x & 31;
    const int half = lane >> 4;      // 0: lanes 0-15, 1: lanes 16-31
    const int l    = lane & 15;
    const int n0   = wave * 16;

    v8f acc[4];
#pragma unroll
    for (int t = 0; t < 4; ++t) acc[t] = (v8f)(0.0f);

#pragma unroll 4
    for (int k0 = 0; k0 < DIM; k0 += 4) {
        const int kk = k0 + 2 * half;

        // B fragment (4x16, K-major per ISA layout): VGPR0 = rows k0 / k0+2,
        // VGPR1 = rows k0+1 / k0+3, N = n0 + (lane & 15).
        v2f b;
        b.x = W[(size_t)(kk    ) * DIM + n0 + l];
        b.y = W[(size_t)(kk + 1) * DIM + n0 + l];

        // A fragments (16x4 f32): lane half selects K-pair {k0,k0+1}/{k0+2,k0+3}.
        v2f a0 = *(const v2f*)(X + (size_t)( 0 + l) * DIM + kk);
        v2f a1 = *(const v2f*)(X + (size_t)(16 + l) * DIM + kk);
        v2f a2 = *(const v2f*)(X + (size_t)(32 + l) * DIM + kk);
        v2f a3 = *(const v2f*)(X + (size_t)(48 + l) * DIM + kk);

        acc[0] = __builtin_amdgcn_wmma_f32_16x16x4_f32(
            false, a0, false, b, (short)0, acc[0], false, false);
        acc[1] = __builtin_amdgcn_wmma_f32_16x16x4_f32(
            false, a1, false, b, (short)0, acc[1], false, false);
        acc[2] = __builtin_amdgcn_wmma_f32_16x16x4_f32(
            false, a2, false, b, (short)0, acc[2], false, false);
        acc[3] = __builtin_amdgcn_wmma_f32_16x16x4_f32(
            false, a3, false, b, (short)0, acc[3], false, false);
    }

    // Bias depends only on the column; add to every accumulator element.
    const float bv = bias[n0 + l];

    // C/D layout: VGPR v holds row (v + 8*half) of the 16x16 tile, col = l.
#pragma unroll
    for (int t = 0; t < 4; ++t) {
#pragma unroll
        for (int v = 0; v < 8; ++v) {
            int row = t * 16 + half * 8 + v;
            if (row < m_store)
                Y[(size_t)row * DIM + n0 + l] = acc[t][v] + bv;
        }
    }
}

extern "C" void kernel_launch(void* const* d_in, const int* in_sizes, int n_in,
                              void* d_out, int out_size, void* d_ws, size_t ws_size,
                              hipStream_t stream) {
    // setup_inputs() insertion order:
    //  [0] comb_fea (unused)  [1] action_fea [1,50,1024]
    //  [2..9]  Wq1 Wk1 Wv1 Wo1 Wq2 Wk2 Wv2 Wo2   (each 1024x1024)
    //  [10..17] bq1 bk1 bv1 bo1 bq2 bk2 bv2 bo2  (each 1024)
    const float* act = (const float*)d_in[1];
    const float* Wv2 = (const float*)d_in[8];
    const float* Wo2 = (const float*)d_in[9];
    const float* bv2 = (const float*)d_in[16];
    const float* bo2 = (const float*)d_in[17];
    float* out = (float*)d_out;

    float* xp = (float*)d_ws;          // 64*1024 f32 = 256 KB
    float* v2 = xp + M_PAD * DIM;      // 64*1024 f32 = 256 KB

    actionprompt_pad_kernel<<<(M_PAD * DIM) / 256, 256, 0, stream>>>(act, xp);
    actionprompt_gemm_wmma_f32<<<8, 256, 0, stream>>>(xp, Wv2, bv2, v2, M_PAD);
    actionprompt_gemm_wmma_f32<<<8, 256, 0, stream>>>(v2, Wo2, bo2, out, A_ROWS);
}